// MyModel_61933428410286
// MI455X (gfx1250) — compile-verified
//
#include <hip/hip_runtime.h>
#include <hip/hip_bf16.h>

// Problem: B=4, S=4096, D=64, fp32.
// out = (Q K^T) V  ==  Q (K^T V)   -> two tiny GEMMs, memory-bound (~0.72us floor).

#define BB 4
#define SS 4096
#define DD 64
#define NCHUNK 16                 // S-chunks per batch for stage-1 parallelism
#define CHUNK (SS / NCHUNK)       // 256 rows per chunk

typedef __attribute__((ext_vector_type(2))) float v2f;
typedef __attribute__((ext_vector_type(8))) float v8f;

// ---------------------------------------------------------------------------
// Stage 1: partial T = K_chunk^T * V_chunk  (64x64 f32), one block per (b,chunk).
// Block = 512 threads = 16 waves; wave w owns output tile (w/4, w%4) of 16x16.
// WMMA f32 16x16x4: A = K^T tile (16 d1-rows x 4 s), B = V tile (4 s x 16 d2).
// ---------------------------------------------------------------------------
__global__ __launch_bounds__(512)
void ktv_partial_kernel(const float* __restrict__ k,
                        const float* __restrict__ v,
                        float* __restrict__ ws_partial) {
    const int b     = blockIdx.x / NCHUNK;
    const int chunk = blockIdx.x % NCHUNK;
    const int s_begin = chunk * CHUNK;

    const int wave = threadIdx.x >> 5;
    const int lane = threadIdx.x & 31;
    const int l    = lane & 15;
    const int hi   = lane >> 4;            // 0: K=0,1   1: K=2,3
    const int kof  = hi << 1;

    const int m_base = (wave >> 2) << 4;   // 0,16,32,48  (d1 tile)
    const int n_base = (wave & 3) << 4;    // 0,16,32,48  (d2 tile)

    const float* kb = k + (size_t)b * SS * DD;
    const float* vb = v + (size_t)b * SS * DD;

    v8f c = {};
    #pragma unroll 4
    for (int s0 = s_begin; s0 < s_begin + CHUNK; s0 += 4) {
        // A[m, kk] = K[s0+kk, m_base+m]   (K^T)
        v2f a;
        a.x = kb[(size_t)(s0 + kof    ) * DD + (m_base + l)];
        a.y = kb[(size_t)(s0 + kof + 1) * DD + (m_base + l)];
        // B[kk, n] = V[s0+kk, n_base+n]
        v2f bmat;
        bmat.x = vb[(size_t)(s0 + kof    ) * DD + (n_base + l)];
        bmat.y = vb[(size_t)(s0 + kof + 1) * DD + (n_base + l)];
        c = __builtin_amdgcn_wmma_f32_16x16x4_f32(
                false, a, false, bmat, (short)0, c, false, false);
    }

    // C/D layout: vgpr r -> row (m_base + r + 8*hi), lanes -> col (n_base + l)
    float* part = ws_partial + (size_t)blockIdx.x * (DD * DD);
    #pragma unroll
    for (int r = 0; r < 8; ++r) {
        const int row = m_base + r + (hi << 3);
        part[(size_t)row * DD + (n_base + l)] = c[r];
    }
}

// ---------------------------------------------------------------------------
// Stage 2: deterministic fixed-order reduction of NCHUNK partials -> T[b] 64x64.
// ---------------------------------------------------------------------------
__global__ __launch_bounds__(256)
void ktv_reduce_kernel(const float* __restrict__ ws_partial,
                       float* __restrict__ t_out) {
    const int b = blockIdx.x;
    for (int idx = threadIdx.x; idx < DD * DD; idx += 256) {
        float acc = 0.0f;
        #pragma unroll
        for (int cidx = 0; cidx < NCHUNK; ++cidx) {
            acc += ws_partial[((size_t)(b * NCHUNK + cidx)) * (DD * DD) + idx];
        }
        t_out[(size_t)b * (DD * DD) + idx] = acc;
    }
}

// ---------------------------------------------------------------------------
// Stage 3: out_tile = Q_tile (64x64) * T (64x64) per block.
// Grid = B * (S/64); block = 512 threads = 16 waves, each wave one 16x16 tile.
// ---------------------------------------------------------------------------
__global__ __launch_bounds__(512)
void qt_gemm_kernel(const float* __restrict__ q,
                    const float* __restrict__ t_in,
                    float* __restrict__ out) {
    const int b        = blockIdx.x / (SS / 64);
    const int rowblock = blockIdx.x % (SS / 64);
    const int row_base = rowblock * 64;

    const int wave = threadIdx.x >> 5;
    const int lane = threadIdx.x & 31;
    const int l    = lane & 15;
    const int hi   = lane >> 4;
    const int kof  = hi << 1;

    const int m_base = (wave >> 2) << 4;   // query-row tile within 64
    const int n_base = (wave & 3) << 4;    // d2 tile

    const float* qb = q    + (size_t)b * SS * DD;
    const float* tb = t_in + (size_t)b * (DD * DD);
    const int row = row_base + m_base + l; // this lane's query row for A

    v8f c = {};
    #pragma unroll
    for (int k0 = 0; k0 < DD; k0 += 4) {
        // A[m, kk] = Q[row, k0+kk]  -> two contiguous floats per lane
        v2f a;
        a.x = qb[(size_t)row * DD + (k0 + kof)];
        a.y = qb[(size_t)row * DD + (k0 + kof + 1)];
        // B[kk, n] = T[k0+kk, n_base+n]
        v2f bmat;
        bmat.x = tb[(size_t)(k0 + kof    ) * DD + (n_base + l)];
        bmat.y = tb[(size_t)(k0 + kof + 1) * DD + (n_base + l)];
        c = __builtin_amdgcn_wmma_f32_16x16x4_f32(
                false, a, false, bmat, (short)0, c, false, false);
    }

    float* ob = out + (size_t)b * SS * DD;
    #pragma unroll
    for (int r = 0; r < 8; ++r) {
        const int orow = row_base + m_base + r + (hi << 3);
        ob[(size_t)orow * DD + (n_base + l)] = c[r];
    }
}

// ---------------------------------------------------------------------------
extern "C" void kernel_launch(void* const* d_in, const int* in_sizes, int n_in,
                              void* d_out, int out_size, void* d_ws, size_t ws_size,
                              hipStream_t stream) {
    const float* q = (const float*)d_in[0];
    const float* k = (const float*)d_in[1];
    const float* v = (const float*)d_in[2];
    float* out = (float*)d_out;

    float* ws_partial = (float*)d_ws;                                   // B*NCHUNK*64*64 floats (1 MB)
    float* t_mat      = ws_partial + (size_t)BB * NCHUNK * DD * DD;     // B*64*64 floats (64 KB)

    ktv_partial_kernel<<<dim3(BB * NCHUNK), dim3(512), 0, stream>>>(k, v, ws_partial);
    ktv_reduce_kernel <<<dim3(BB),          dim3(256), 0, stream>>>(ws_partial, t_mat);
    qt_gemm_kernel    <<<dim3(BB * (SS/64)), dim3(512), 0, stream>>>(q, t_mat, out);
}